// CDAATRACK_43585328120132
// MI455X (gfx1250) — compile-verified
//
#include <hip/hip_runtime.h>

// ---------------------------------------------------------------------------
// CDAATRACK priors: color/depth histogram build + per-pixel prior lookup.
// Pass 1: per-batch LDS histograms (ds atomics); normalized [fore,back] pairs
//         written to d_ws (L2-resident, 2.2 MB total).
// Pass 2: async global->LDS staging of the pair tables, ds_load_b64 gathers,
//         non-temporal streaming loads/stores for the 300 MB of one-shot data.
// ---------------------------------------------------------------------------

#define B_      64
#define HT      128
#define WT      128
#define TPX     (HT * WT)            // 16384 template pixels
#define HS      384
#define WSX     384
#define SPX     (HS * WSX)           // 147456 search pixels
#define NBC     4096                 // 16^3 color bins
#define NBD     256                  // depth bins
#define NPAIR   (NBC + NBD)          // 4352 float2 pairs per batch
#define HIST_F  (2 * NPAIR)          // 8704 floats per batch in scratch
#define EPS_    1e-5f

#define THREADS 256
#define PBLK    48                   // blocks per batch in pass 2
#define CHUNK   (SPX / PBLK)         // 3072 pixels per block

// ---------------------------------------------------------------------------
// Pass 1: histograms. One block per batch image.
// ---------------------------------------------------------------------------
__global__ __launch_bounds__(THREADS) void cdaa_hist_kernel(
    const float* __restrict__ anno,   // [B,4] xywh
    const int*   __restrict__ tcol,   // [B,HT,WT,3]
    const int*   __restrict__ tdep,   // [B,HT,WT]
    float*       __restrict__ ws)     // [B, NPAIR] float2 {fore,back} pairs
{
    __shared__ unsigned s_fc[NBC];
    __shared__ unsigned s_bc[NBC];
    __shared__ unsigned s_fd[NBD];
    __shared__ unsigned s_bd[NBD];
    __shared__ unsigned s_cnt;

    const int b   = blockIdx.x;
    const int tid = threadIdx.x;

    for (int i = tid; i < NBC; i += THREADS) { s_fc[i] = 0u; s_bc[i] = 0u; }
    for (int i = tid; i < NBD; i += THREADS) { s_fd[i] = 0u; s_bd[i] = 0u; }
    if (tid == 0) s_cnt = 0u;
    __syncthreads();

    // jnp.round == round-half-to-even == rintf (default rounding mode)
    const float bx = rintf(anno[b * 4 + 0]);
    const float by = rintf(anno[b * 4 + 1]);
    const float bw = rintf(anno[b * 4 + 2]);
    const float bh = rintf(anno[b * 4 + 3]);
    const int xmin = (int)bx, xmax = (int)(bx + bw);
    const int ymin = (int)by, ymax = (int)(by + bh);

    const int* tc = tcol + (size_t)b * TPX * 3;
    const int* td = tdep + (size_t)b * TPX;

    unsigned cnt = 0u;
    for (int p = tid; p < TPX; p += THREADS) {
        const int y  = p >> 7;
        const int x  = p & (WT - 1);
        const int r  = __builtin_nontemporal_load(tc + p * 3 + 0);
        const int g  = __builtin_nontemporal_load(tc + p * 3 + 1);
        const int bl = __builtin_nontemporal_load(tc + p * 3 + 2);
        const int bin = ((r >> 4) << 8) | ((g >> 4) << 4) | (bl >> 4);
        const int d   = __builtin_nontemporal_load(td + p) & 255;
        const bool fore = (y >= ymin) & (y < ymax) & (x >= xmin) & (x < xmax);
        if (fore) {
            atomicAdd(&s_fc[bin], 1u);
            atomicAdd(&s_fd[d], 1u);
            cnt++;
        } else {
            atomicAdd(&s_bc[bin], 1u);
            atomicAdd(&s_bd[d], 1u);
        }
    }
    atomicAdd(&s_cnt, cnt);
    __syncthreads();

    const unsigned nf = s_cnt;
    const float invf = 1.0f / (float)(nf + 1u);
    const float invb = 1.0f / (float)((unsigned)TPX - nf + 1u);

    // Interleaved {fore,back} pairs -> one ds_load_b64 per gather in pass 2.
    float2* wp = (float2*)(ws + (size_t)b * HIST_F);
    for (int i = tid; i < NBC; i += THREADS)
        wp[i] = make_float2((float)s_fc[i] * invf, (float)s_bc[i] * invb);
    for (int i = tid; i < NBD; i += THREADS)
        wp[NBC + i] = make_float2((float)s_fd[i] * invf, (float)s_bd[i] * invb);
}

// ---------------------------------------------------------------------------
// Pass 2: per-pixel prior lookup. grid = (B, PBLK).
// Histogram pair tables staged into LDS via gfx1250 async global->LDS loads.
// Builtin signature (probe-confirmed): (v4i AS1*, v4i AS3*, Ii offset, Ii cpol)
// ---------------------------------------------------------------------------
typedef int v4i __attribute__((ext_vector_type(4)));
typedef __attribute__((address_space(1))) v4i* gv4i_p;
typedef __attribute__((address_space(3))) v4i* lv4i_p;

#if defined(__has_builtin)
#if __has_builtin(__builtin_amdgcn_global_load_async_to_lds_b128)
#define HAVE_ASYNC_BUILTIN 1
#endif
#endif

__global__ __launch_bounds__(THREADS) void cdaa_prior_kernel(
    const int*   __restrict__ scol,   // [B,HS,WS,3]
    const int*   __restrict__ sdep,   // [B,HS,WS]
    const float* __restrict__ ws,     // [B, NPAIR] float2 pairs
    float*       __restrict__ out)    // [B,4,HS,WS]
{
    __shared__ float2 s_p[NPAIR];     // [color pairs | depth pairs], 34 KB

    const int b   = blockIdx.x;
    const int tid = threadIdx.x;
    const float* w = ws + (size_t)b * HIST_F;
    float* s_h = (float*)s_p;

    // ---- stage 34 KB of histogram pairs into LDS (async, b128 granules) ----
#ifdef HAVE_ASYNC_BUILTIN
    for (int i = tid; i < HIST_F / 4; i += THREADS) {
        __builtin_amdgcn_global_load_async_to_lds_b128(
            (gv4i_p)(w + i * 4), (lv4i_p)(&s_h[i * 4]), 0, 0);
    }
#else
    for (int i = tid; i < HIST_F / 4; i += THREADS) {
        unsigned lds_addr = (unsigned)(__UINTPTR_TYPE__)(
            (__attribute__((address_space(3))) void*)&s_h[i * 4]);
        unsigned long long gaddr =
            (unsigned long long)(__UINTPTR_TYPE__)(w + i * 4);
        asm volatile("global_load_async_to_lds_b128 %0, %1, off"
                     :: "v"(lds_addr), "v"(gaddr) : "memory");
    }
#endif
#if defined(__has_builtin) && __has_builtin(__builtin_amdgcn_s_wait_asynccnt)
    __builtin_amdgcn_s_wait_asynccnt(0);
#else
    asm volatile("s_wait_asynccnt 0" ::: "memory");
#endif
    __syncthreads();

    // ---- per-pixel priors (streaming NT in, NT out; tables stay hot in L2) --
    const int p0 = blockIdx.y * CHUNK;
    const int* sc = scol + (size_t)b * SPX * 3;
    const int* sd = sdep + (size_t)b * SPX;
    float* ob = out + (size_t)b * 4 * SPX;

    for (int k = tid; k < CHUNK; k += THREADS) {
        const int p  = p0 + k;
        const int r  = __builtin_nontemporal_load(sc + p * 3 + 0);
        const int g  = __builtin_nontemporal_load(sc + p * 3 + 1);
        const int bl = __builtin_nontemporal_load(sc + p * 3 + 2);
        const int bin = ((r >> 4) << 8) | ((g >> 4) << 4) | (bl >> 4);
        const int d   = __builtin_nontemporal_load(sd + p) & 255;

        const float2 pc = s_p[bin];         // ds_load_b64: {fore_c, back_c}
        const float2 pd = s_p[NBC + d];     // ds_load_b64: {fore_d, back_d}

        const float denc = pc.x + pc.y + EPS_;
        const float dend = pd.x + pd.y + EPS_;

        __builtin_nontemporal_store(pc.x / denc, ob + p);            // c fore
        __builtin_nontemporal_store(pc.y / denc, ob + SPX + p);      // c back
        __builtin_nontemporal_store(pd.x / dend, ob + 2 * SPX + p);  // d fore
        __builtin_nontemporal_store(pd.y / dend, ob + 3 * SPX + p);  // d back
    }
}

// ---------------------------------------------------------------------------
// Launcher
// ---------------------------------------------------------------------------
extern "C" void kernel_launch(void* const* d_in, const int* in_sizes, int n_in,
                              void* d_out, int out_size, void* d_ws, size_t ws_size,
                              hipStream_t stream) {
    const float* anno = (const float*)d_in[0];  // target_anno   [B,4]
    const int*   tcol = (const int*)d_in[1];    // template_color[B,128,128,3]
    const int*   scol = (const int*)d_in[2];    // search_color  [B,384,384,3]
    const int*   tdep = (const int*)d_in[3];    // template_depth[B,128,128]
    const int*   sdep = (const int*)d_in[4];    // search_depth  [B,384,384]
    float* out = (float*)d_out;                 // [B,4,384,384] f32
    float* ws  = (float*)d_ws;                  // needs B*HIST_F*4 = ~2.23 MB

    (void)in_sizes; (void)n_in; (void)out_size; (void)ws_size;

    cdaa_hist_kernel<<<dim3(B_), dim3(THREADS), 0, stream>>>(anno, tcol, tdep, ws);
    cdaa_prior_kernel<<<dim3(B_, PBLK), dim3(THREADS), 0, stream>>>(scol, sdep, ws, out);
}